// ScaleManagementExample_58128087384357
// MI455X (gfx1250) — compile-verified
//
#include <hip/hip_runtime.h>
#include <stdint.h>

// Problem constants (fixed by the reference)
#define NB     8
#define CIN    3
#define HH     128
#define WW     128
#define OC     16
#define NF     27                        // CIN*3*3, ordered (cin, kh, kw)
#define NPIX_IN  (NB*CIN*HH*WW)          // 393216 activation elements
#define NPIX_OUT (NB*HH*WW)              // 131072 output pixels
#define NROWS   (NF*256)                 // 6912 (f,a) rows
#define PLANE_SHORTS (NROWS*8)           // 55296 int16 per half-plane
#define PLANE_BYTES  (PLANE_SHORTS*2)    // 110592 bytes
#define PTAB_ELEMS   (NROWS*OC)          // 110592 int16 entries total
#define PIX_PER_THREAD 2

#define CDNA5_STR2(x) #x
#define CDNA5_STR(x) CDNA5_STR2(x)
#pragma message("CDNA5 probe: __clang_major__=" CDNA5_STR(__clang_major__))

#if __has_builtin(__builtin_amdgcn_global_load_async_to_lds_b128) && \
    __has_builtin(__builtin_amdgcn_s_wait_asynccnt)
#define USE_ASYNC_LDS 1
typedef int v4i __attribute__((ext_vector_type(4)));
typedef __attribute__((address_space(1))) v4i gv4i;   // global (cuda_device)
typedef __attribute__((address_space(3))) v4i lv4i;   // LDS
#endif

#if __has_builtin(__builtin_amdgcn_tensor_load_to_lds)
#pragma message("CDNA5 probe: has __builtin_amdgcn_tensor_load_to_lds")
#else
#pragma message("CDNA5 probe: NO __builtin_amdgcn_tensor_load_to_lds")
#endif

// TDM path: clang<=22 (ROCm 7.2) uses the 5-arg form; clang>=23 the 6-arg
// form whose extra i32x8 matches the architecturally-unused VADDR4 (zeroed).
#if __has_builtin(__builtin_amdgcn_tensor_load_to_lds) && \
    __has_builtin(__builtin_amdgcn_s_wait_tensorcnt)
#define USE_TDM 1
typedef unsigned int u32x4 __attribute__((ext_vector_type(4)));
typedef int i32x8 __attribute__((ext_vector_type(8)));
typedef int i32x4 __attribute__((ext_vector_type(4)));
#if defined(__clang_major__) && (__clang_major__ >= 23)
#define TDM_LOAD(g0, g1, g2, g3) \
  __builtin_amdgcn_tensor_load_to_lds((g0), (g1), (g2), (g3), \
                                      (i32x8){0,0,0,0,0,0,0,0}, 0)
#else
#define TDM_LOAD(g0, g1, g2, g3) \
  __builtin_amdgcn_tensor_load_to_lds((g0), (g1), (g2), (g3), 0)
#endif
#endif

#if defined(USE_TDM)
#pragma message("CDNA5 probe: fill path = TDM tensor_load_to_lds")
#elif defined(USE_ASYNC_LDS)
#pragma message("CDNA5 probe: fill path = global_load_async_to_lds_b128")
#else
#pragma message("CDNA5 probe: fill path = plain loads (fallback)")
#endif

// ---------------------------------------------------------------------------
// Kernel 1: quantize activations to biased uint8 ip = clip(round(x/sx)) + 128
// jnp.round is round-half-to-even -> rintf (default RNE mode); exact IEEE div.
// ---------------------------------------------------------------------------
__global__ __launch_bounds__(256) void lutconv_quant_x(
    const float* __restrict__ x, const float* __restrict__ scale_x,
    uint8_t* __restrict__ ip, int n)
{
  int i = blockIdx.x * blockDim.x + threadIdx.x;
  if (i >= n) return;
  float s = scale_x[0];
  float q = rintf(x[i] / s);
  q = fminf(fmaxf(q, -127.0f), 127.0f);
  ip[i] = (uint8_t)((int)q + 128);
}

// ---------------------------------------------------------------------------
// Kernel 2: pre-contract weights into the LUT:
//   value(f,a,o) = T[a*256 + (clip(round(w[o,f]/sw))+128)],  T = hi*256+(lo&255)
// Stored as two planes of 16B rows (o 0-7 -> plane A, o 8-15 -> plane B) so the
// LDS gather's bank-group start becomes addr[7:4]=a[3:0] (16 groups, not 8).
// ---------------------------------------------------------------------------
__global__ __launch_bounds__(256) void lutconv_build_ptab(
    const float* __restrict__ weight, const float* __restrict__ scale_w,
    const int* __restrict__ lut, short* __restrict__ ptab)
{
  int e = blockIdx.x * blockDim.x + threadIdx.x;
  if (e >= PTAB_ELEMS) return;
  int o = e & (OC - 1);
  int a = (e >> 4) & 255;
  int f = e >> 12;
  float s = scale_w[0];
  float q = rintf(weight[o * NF + f] / s);
  q = fminf(fmaxf(q, -127.0f), 127.0f);
  int iw  = (int)q + 128;
  int idx = a * 256 + iw;
  int hi = lut[idx * 2];
  int lo = lut[idx * 2 + 1] & 255;
  int r = (f << 8) | a;
  int dst = (o < 8) ? (r * 8 + o) : (PLANE_SHORTS + r * 8 + (o - 8));
  ptab[dst] = (short)(hi * 256 + lo);
}

// ---------------------------------------------------------------------------
// Kernel 3: LUT-approximated 3x3 conv. Both ptab planes (216KB total) live in
// LDS (CDNA5: 320KB/workgroup). Fill via TDM tensor_load_to_lds (one DMA per
// plane) when available, else GLOBAL_LOAD_ASYNC_TO_LDS_B128, else plain loads.
// Per pixel: 27 taps x (2x ds_load_b128 gather) into 16 int32 accumulators.
// Padding taps use a = 128 (quantized 0 + 128); the reference sums them too.
// ---------------------------------------------------------------------------
#define ACC_DWORD(A0, A1, D) { A0 += (int)(short)((D) & 0xFFFF); A1 += ((D) >> 16); }

__global__ __launch_bounds__(256) void lutconv_main(
    const uint8_t* __restrict__ ip, const short* __restrict__ ptab,
    const float* __restrict__ bias, const float* __restrict__ scale_x,
    const float* __restrict__ scale_w, float* __restrict__ out)
{
  __shared__ short lA[PLANE_SHORTS];   // 110592 bytes, o = 0..7
  __shared__ short lB[PLANE_SHORTS];   // 110592 bytes, o = 8..15
  const int tid = threadIdx.x;

#if defined(USE_TDM)
  if (tid < 32) {                       // wave 0 only issues the two DMAs
    uint64_t gaA = (uint64_t)(uintptr_t)ptab;
    uint64_t gaB = gaA + PLANE_BYTES;
    uint32_t ldsA = (uint32_t)(uintptr_t)lA;
    uint32_t ldsB = (uint32_t)(uintptr_t)lB;
    // D# group1: data_size=2B (code 1); 1-D tile: tensor_dim0=tile_dim0=55296,
    // tensor_dim1=1, tile_dim1=0 (unused); stride0=55296; no pad/iterate/mask.
    i32x8 g1;
    g1[0] = (1 << 16);
    g1[1] = (int)((PLANE_SHORTS & 0xFFFF) << 16);         // tensor_dim0 lo16
    g1[2] = (1 << 16) | (PLANE_SHORTS >> 16);             // tensor_dim1 lo16 | dim0 hi16
    g1[3] = (int)((PLANE_SHORTS & 0xFFFF) << 16);         // tile_dim0 | tensor_dim1 hi16
    g1[4] = 0;                                            // tile_dim2 | tile_dim1 (unused)
    g1[5] = PLANE_SHORTS;                                 // tensor_dim0_stride lo32
    g1[6] = 0;
    g1[7] = 0;
    i32x4 gz = {0, 0, 0, 0};
    u32x4 g0A = { 1u, ldsA, (uint32_t)gaA,
                  (uint32_t)(((gaA >> 32) & 0x1FFFFFFu) | (2u << 30)) };
    u32x4 g0B = { 1u, ldsB, (uint32_t)gaB,
                  (uint32_t)(((gaB >> 32) & 0x1FFFFFFu) | (2u << 30)) };
    TDM_LOAD(g0A, g1, gz, gz);
    TDM_LOAD(g0B, g1, gz, gz);
    __builtin_amdgcn_s_wait_tensorcnt(0);
  }
#elif defined(USE_ASYNC_LDS)
  {
    const char* gA = (const char*)ptab;
    const char* gB = gA + PLANE_BYTES;
    char* sA = (char*)lA;
    char* sB = (char*)lB;
    for (int ofs = tid * 16; ofs < PLANE_BYTES; ofs += 256 * 16) {
      __builtin_amdgcn_global_load_async_to_lds_b128(
          (gv4i*)(uintptr_t)(gA + ofs), (lv4i*)(uint32_t)(uintptr_t)(sA + ofs), 0, 0);
      __builtin_amdgcn_global_load_async_to_lds_b128(
          (gv4i*)(uintptr_t)(gB + ofs), (lv4i*)(uint32_t)(uintptr_t)(sB + ofs), 0, 0);
    }
    __builtin_amdgcn_s_wait_asynccnt(0);
  }
#else
  for (int e = tid; e < PLANE_SHORTS; e += 256) { lA[e] = ptab[e]; lB[e] = ptab[PLANE_SHORTS + e]; }
#endif
  __syncthreads();

  const float sc = scale_x[0] * scale_w[0];

  #pragma unroll
  for (int it = 0; it < PIX_PER_THREAD; ++it) {
    int p = blockIdx.x * (256 * PIX_PER_THREAD) + it * 256 + tid;
    if (p >= NPIX_OUT) continue;
    int b = p >> 14;
    int rem = p & 16383;
    int y = rem >> 7;
    int x = rem & 127;

    int acc[OC];
    #pragma unroll
    for (int o = 0; o < OC; ++o) acc[o] = 0;

    #pragma unroll
    for (int cin = 0; cin < CIN; ++cin) {
      int base = (b * CIN + cin) << 14;
      #pragma unroll
      for (int dy = 0; dy < 3; ++dy) {
        int yy = y + dy - 1;
        bool rowok = (unsigned)yy < (unsigned)HH;
        #pragma unroll
        for (int dx = 0; dx < 3; ++dx) {
          int xx = x + dx - 1;
          int a = 128;  // padded tap: quantized 0 -> index 128
          if (rowok && (unsigned)xx < (unsigned)WW) a = ip[base + (yy << 7) + xx];
          int f = cin * 9 + dy * 3 + dx;
          int r = (f << 8) | a;
          int4 v0 = *(const int4*)&lA[r << 3];
          int4 v1 = *(const int4*)&lB[r << 3];
          ACC_DWORD(acc[0],  acc[1],  v0.x);
          ACC_DWORD(acc[2],  acc[3],  v0.y);
          ACC_DWORD(acc[4],  acc[5],  v0.z);
          ACC_DWORD(acc[6],  acc[7],  v0.w);
          ACC_DWORD(acc[8],  acc[9],  v1.x);
          ACC_DWORD(acc[10], acc[11], v1.y);
          ACC_DWORD(acc[12], acc[13], v1.z);
          ACC_DWORD(acc[14], acc[15], v1.w);
        }
      }
    }

    int obase = (b << 18) | (y << 7) | x;   // b*OC*16384
    #pragma unroll
    for (int o = 0; o < OC; ++o)
      out[obase + (o << 14)] = (float)acc[o] * sc + bias[o];
  }
}

// ---------------------------------------------------------------------------
// Launch. Workspace layout (needs 614400 bytes):
//   [0, 393216)          ip   (uint8, biased quantized activations)
//   [393216, 614400)     ptab (int16: plane A 110592B, then plane B 110592B)
// ---------------------------------------------------------------------------
extern "C" void kernel_launch(void* const* d_in, const int* in_sizes, int n_in,
                              void* d_out, int out_size, void* d_ws, size_t ws_size,
                              hipStream_t stream) {
  const float* x      = (const float*)d_in[0];
  const float* weight = (const float*)d_in[1];
  const float* bias   = (const float*)d_in[2];
  const float* sx     = (const float*)d_in[3];
  const float* sw     = (const float*)d_in[4];
  const int*   lut    = (const int*)d_in[5];
  float* out = (float*)d_out;

  uint8_t* ip   = (uint8_t*)d_ws;
  short*   ptab = (short*)((char*)d_ws + NPIX_IN);  // 393216 is 256B aligned

  lutconv_quant_x<<<(NPIX_IN + 255) / 256, 256, 0, stream>>>(x, sx, ip, NPIX_IN);
  lutconv_build_ptab<<<(PTAB_ELEMS + 255) / 256, 256, 0, stream>>>(weight, sw, lut, ptab);
  lutconv_main<<<NPIX_OUT / (256 * PIX_PER_THREAD), 256, 0, stream>>>(
      ip, ptab, bias, sx, sw, out);
}